// QNN_41695542510114
// MI455X (gfx1250) — compile-verified
//
#include <hip/hip_runtime.h>
#include <hip/hip_bf16.h>
#include <math.h>

// ---------------------------------------------------------------------------
// Problem constants (from reference)
// ---------------------------------------------------------------------------
#define NQ    14
#define DIM   16384          // 2^14
#define OUT_N 1000
#define BATCH 512
#define NPAD  1024           // N padded to 64 tiles of 16

typedef __attribute__((ext_vector_type(2))) float v2f;
typedef __attribute__((ext_vector_type(8))) float v8f;

// Persistent scratch (recomputed every call; deterministic).
__device__ float g_W[(size_t)NPAD * DIM];   // W[j][k] = U[DIM-OUT_N+j, k] (rows >= OUT_N zero)
__device__ float g_inv_norm[BATCH];

// ---------------------------------------------------------------------------
// Kernel 1: per-row inverse norms of X
// ---------------------------------------------------------------------------
__global__ void __launch_bounds__(256) qnn_rownorm(const float* __restrict__ X) {
    __shared__ float red[256];
    const int b = blockIdx.x;
    const int tid = threadIdx.x;
    const float* row = X + (size_t)b * DIM;
    float acc = 0.0f;
    for (int i = tid; i < DIM; i += 256) {
        float v = row[i];
        acc += v * v;
    }
    red[tid] = acc;
    __syncthreads();
    for (int s = 128; s > 0; s >>= 1) {
        if (tid < s) red[tid] += red[tid + s];
        __syncthreads();
    }
    if (tid == 0) g_inv_norm[b] = rsqrtf(red[0]);
}

// ---------------------------------------------------------------------------
// Kernel 2: build W rows by simulating the transposed circuit on basis vectors.
// State (64 KB) lives in LDS. qubit q <-> bit (13-q) of the flat index.
// ---------------------------------------------------------------------------
__device__ __forceinline__ void apply_ry_lds(float* st, int q, float ang, int tid) {
    const int s = 1 << (13 - q);
    float sn, cs;
    sincosf(0.5f * ang, &sn, &cs);
    for (int t = tid; t < (DIM / 2); t += 256) {
        const int low  = t & (s - 1);
        const int high = (t & ~(s - 1)) << 1;
        const int i0 = high | low;
        const int i1 = i0 | s;
        const float x0 = st[i0];
        const float x1 = st[i1];
        st[i0] = cs * x0 - sn * x1;
        st[i1] = sn * x0 + cs * x1;
    }
    __syncthreads();
}

__device__ __forceinline__ void apply_cz_lds(float* st, int q, int tid) {
    const int b0 = 1 << (13 - q);
    const int b1 = 1 << (13 - (q + 1));
    const int both = b0 | b1;
    for (int i = tid; i < DIM; i += 256) {
        if ((i & both) == both) st[i] = -st[i];
    }
    __syncthreads();
}

__global__ void __launch_bounds__(256) qnn_build_w(const float* __restrict__ wts,
                                                   const float* __restrict__ bias) {
    __shared__ float st[DIM];     // 64 KB
    const int j = blockIdx.x;     // padded row index, 0..NPAD-1
    const int tid = threadIdx.x;

    for (int i = tid; i < DIM; i += 256) st[i] = 0.0f;
    __syncthreads();

    if (j < OUT_N) {
        if (tid == 0) st[DIM - OUT_N + j] = 1.0f;
        __syncthreads();

        // U^T = reversed gate sequence, RY(t)->RY(-t), CZ unchanged.
        for (int l = 1; l >= 0; --l) {
            for (int start = 1; start >= 0; --start) {
                const int ilast = (start == 0) ? 12 : 11;
                for (int i = ilast; i >= start; i -= 2) {
                    apply_ry_lds(st, i + 1, -wts[(l * (NQ - 1) + i) * 2 + 1], tid);
                    apply_ry_lds(st, i,     -wts[(l * (NQ - 1) + i) * 2 + 0], tid);
                    apply_cz_lds(st, i, tid);
                }
            }
        }
        for (int q = NQ - 1; q >= 0; --q) {
            apply_ry_lds(st, q, -bias[q], tid);
        }
    }

    float* row = g_W + (size_t)j * DIM;
    for (int i = tid; i < DIM; i += 256) row[i] = st[i];
}

// ---------------------------------------------------------------------------
// Kernel 3: out[b, j] = ( dot(X[b,:], W[j,:]) * inv_norm[b] )^2
// WMMA f32 16x16x4 GEMM.
//   Block tile 64(M) x 128(N), K-block 32, 8 waves arranged 2(M) x 4(N),
//   each wave owns a 32x32 register tile (4x v8f accumulators):
//     per 4-K step -> 4 DS pair-loads feed 4 WMMAs (1 DS op / WMMA).
//   LDS is double-buffered; global->reg prefetch overlaps compute.
// ---------------------------------------------------------------------------
#define BM 64
#define BN 128
#define BK 32

__device__ __forceinline__ v8f wmma_f32(v2f a, v2f b, v8f c) {
    return __builtin_amdgcn_wmma_f32_16x16x4_f32(
        false, a, false, b, (short)0, c, false, false);
}

__global__ void __launch_bounds__(256) qnn_gemm_wmma(const float* __restrict__ X,
                                                     float* __restrict__ out) {
    __shared__ float Xs[2][BM][BK + 1];   // 2 * 64 * 33 * 4  = 16.9 KB
    __shared__ float Ws[2][BN][BK + 1];   // 2 * 128 * 33 * 4 = 33.8 KB

    const int tid  = threadIdx.x;
    const int lane = tid & 31;
    const int wave = tid >> 5;       // 0..7
    const int wm   = wave & 1;       // 2 waves along M
    const int wn   = wave >> 1;      // 4 waves along N
    const int m0   = blockIdx.x * BM;    // 8 blocks in M
    const int n0   = blockIdx.y * BN;    // 8 blocks in N

    v8f c00 = {}, c01 = {}, c10 = {}, c11 = {};

    // Operand layout for 32-bit WMMA (16x4 A / 4x16 B):
    //   lanes 0-15: {K, K+1}; lanes 16-31: {K+2, K+3}; M/N = lane&15.
    const int mr = wm * 32 + (lane & 15);
    const int nr = wn * 32 + (lane & 15);
    const int kb = (lane >> 4) * 2;

    // Cooperative loader geometry (float4 granularity):
    //   X tile 64x32  = 512 float4 -> 2 per thread
    //   W tile 128x32 = 1024 float4 -> 4 per thread
    const int xr_row[2] = { (tid + 0)   >> 3, (tid + 256) >> 3 };
    const int xr_col    = (tid & 7) * 4;
    float4 xbuf[2];
    float4 wbuf[4];

    // ---- prologue: stage K-block 0 into buffer 0 ----
    #pragma unroll
    for (int it = 0; it < 2; ++it)
        xbuf[it] = *(const float4*)(X + (size_t)(m0 + xr_row[it]) * DIM + xr_col);
    #pragma unroll
    for (int it = 0; it < 4; ++it) {
        const int f = tid + it * 256;
        wbuf[it] = *(const float4*)(g_W + (size_t)(n0 + (f >> 3)) * DIM + (f & 7) * 4);
    }
    #pragma unroll
    for (int it = 0; it < 2; ++it) {
        float4 v = xbuf[it];
        Xs[0][xr_row[it]][xr_col + 0] = v.x; Xs[0][xr_row[it]][xr_col + 1] = v.y;
        Xs[0][xr_row[it]][xr_col + 2] = v.z; Xs[0][xr_row[it]][xr_col + 3] = v.w;
    }
    #pragma unroll
    for (int it = 0; it < 4; ++it) {
        const int f = tid + it * 256;
        const int r = f >> 3, cc = (f & 7) * 4;
        float4 v = wbuf[it];
        Ws[0][r][cc + 0] = v.x; Ws[0][r][cc + 1] = v.y;
        Ws[0][r][cc + 2] = v.z; Ws[0][r][cc + 3] = v.w;
    }
    __syncthreads();

    int p = 0;
    for (int k0 = 0; k0 < DIM; k0 += BK) {
        const int kn = k0 + BK;
        const bool has_next = (kn < DIM);

        // Prefetch next K-block into registers (overlaps WMMA compute).
        if (has_next) {
            #pragma unroll
            for (int it = 0; it < 2; ++it)
                xbuf[it] = *(const float4*)(X + (size_t)(m0 + xr_row[it]) * DIM + kn + xr_col);
            #pragma unroll
            for (int it = 0; it < 4; ++it) {
                const int f = tid + it * 256;
                wbuf[it] = *(const float4*)(g_W + (size_t)(n0 + (f >> 3)) * DIM + kn + (f & 7) * 4);
            }
        }

        // Compute on buffer p: 8 K-steps x 4 WMMAs.
        #pragma unroll
        for (int kk = 0; kk < BK; kk += 4) {
            v2f a0, a1, b0, b1;
            a0.x = Xs[p][mr][kk + kb];       a0.y = Xs[p][mr][kk + kb + 1];
            a1.x = Xs[p][mr + 16][kk + kb];  a1.y = Xs[p][mr + 16][kk + kb + 1];
            b0.x = Ws[p][nr][kk + kb];       b0.y = Ws[p][nr][kk + kb + 1];
            b1.x = Ws[p][nr + 16][kk + kb];  b1.y = Ws[p][nr + 16][kk + kb + 1];
            c00 = wmma_f32(a0, b0, c00);
            c01 = wmma_f32(a0, b1, c01);
            c10 = wmma_f32(a1, b0, c10);
            c11 = wmma_f32(a1, b1, c11);
        }

        // Drain prefetch into the other buffer.
        if (has_next) {
            const int q = p ^ 1;
            #pragma unroll
            for (int it = 0; it < 2; ++it) {
                float4 v = xbuf[it];
                Xs[q][xr_row[it]][xr_col + 0] = v.x; Xs[q][xr_row[it]][xr_col + 1] = v.y;
                Xs[q][xr_row[it]][xr_col + 2] = v.z; Xs[q][xr_row[it]][xr_col + 3] = v.w;
            }
            #pragma unroll
            for (int it = 0; it < 4; ++it) {
                const int f = tid + it * 256;
                const int r = f >> 3, cc = (f & 7) * 4;
                float4 v = wbuf[it];
                Ws[q][r][cc + 0] = v.x; Ws[q][r][cc + 1] = v.y;
                Ws[q][r][cc + 2] = v.z; Ws[q][r][cc + 3] = v.w;
            }
        }
        __syncthreads();
        p ^= 1;
    }

    // C/D layout: VGPR v -> M = v + 8*(lane>>4); N = lane&15. Scale, square, store.
    #pragma unroll
    for (int ni = 0; ni < 2; ++ni) {
        const int n = n0 + wn * 32 + ni * 16 + (lane & 15);
        if (n < OUT_N) {
            #pragma unroll
            for (int mi = 0; mi < 2; ++mi) {
                const v8f cc = (ni == 0) ? (mi == 0 ? c00 : c10)
                                         : (mi == 0 ? c01 : c11);
                #pragma unroll
                for (int v = 0; v < 8; ++v) {
                    const int m = m0 + wm * 32 + mi * 16 + v + ((lane >> 4) * 8);
                    const float val = cc[v] * g_inv_norm[m];
                    out[(size_t)m * OUT_N + n] = val * val;
                }
            }
        }
    }
}

// ---------------------------------------------------------------------------
// Launch
// ---------------------------------------------------------------------------
extern "C" void kernel_launch(void* const* d_in, const int* in_sizes, int n_in,
                              void* d_out, int out_size, void* d_ws, size_t ws_size,
                              hipStream_t stream) {
    (void)in_sizes; (void)n_in; (void)d_ws; (void)ws_size; (void)out_size;

    const float* X    = (const float*)d_in[0];   // (512, 16384) f32
    const float* wts  = (const float*)d_in[1];   // (2, 13, 2)   f32
    const float* bias = (const float*)d_in[2];   // (14,)        f32
    float* out = (float*)d_out;                  // (512, 1000)  f32

    qnn_rownorm<<<BATCH, 256, 0, stream>>>(X);
    qnn_build_w<<<NPAD, 256, 0, stream>>>(wts, bias);

    dim3 grid(BATCH / BM, NPAD / BN);            // (8, 8)
    qnn_gemm_wmma<<<grid, 256, 0, stream>>>(X, out);
}